// LDRLoss_56160992362590
// MI455X (gfx1250) — compile-verified
//
#include <hip/hip_runtime.h>
#include <math.h>

// Problem geometry (fixed by the reference)
#define B_ROWS 32
#define T_LEN  262144
#define N_TOT  (B_ROWS * T_LEN)        // 8388608 = 2^23
#define SEG    512
#define SPR    (T_LEN / SEG)           // 512 segments per row
#define NSEG   (B_ROWS * SPR)          // 16384 segments total
#define SHIFT  65047                   // flat circular roll amount

typedef float v2f __attribute__((ext_vector_type(2)));
typedef float v8f __attribute__((ext_vector_type(8)));

// ---------------------------------------------------------------------------
// Phase 1: per-segment zero-init EMA scans (4 chains: short/long x pred/targ).
// Each thread owns one contiguous 512-sample segment of one row and emits the
// segment's final state ("B" aggregate). Exact-scan decomposition: the
// segment transform is y_out = A * y_in + B with A = (1-c)^SEG (constant).
// ---------------------------------------------------------------------------
__global__ void __launch_bounds__(256)
ldr_phase1_segscan(const float* __restrict__ pred,
                   const float* __restrict__ targ,
                   float* __restrict__ bvals,   // [4][NSEG]
                   float cs, float cms, float cl, float cml)
{
    const int seg = blockIdx.x * blockDim.x + threadIdx.x;   // 0..NSEG-1
    const int row = seg / SPR;
    const int s   = seg % SPR;
    const size_t base = (size_t)row * T_LEN + (size_t)s * SEG;

    const float4* __restrict__ p4 = reinterpret_cast<const float4*>(pred + base);
    const float4* __restrict__ t4 = reinterpret_cast<const float4*>(targ + base);

    float ysp = 0.f, ylp = 0.f, yst = 0.f, ylt = 0.f;
#pragma unroll 4
    for (int i = 0; i < SEG / 4; ++i) {
        float4 xp = p4[i];
        float4 xt = t4[i];
        ysp = fmaf(cms, ysp, cs * xp.x); ylp = fmaf(cml, ylp, cl * xp.x);
        yst = fmaf(cms, yst, cs * xt.x); ylt = fmaf(cml, ylt, cl * xt.x);
        ysp = fmaf(cms, ysp, cs * xp.y); ylp = fmaf(cml, ylp, cl * xp.y);
        yst = fmaf(cms, yst, cs * xt.y); ylt = fmaf(cml, ylt, cl * xt.y);
        ysp = fmaf(cms, ysp, cs * xp.z); ylp = fmaf(cml, ylp, cl * xp.z);
        yst = fmaf(cms, yst, cs * xt.z); ylt = fmaf(cml, ylt, cl * xt.z);
        ysp = fmaf(cms, ysp, cs * xp.w); ylp = fmaf(cml, ylp, cl * xp.w);
        yst = fmaf(cms, yst, cs * xt.w); ylt = fmaf(cml, ylt, cl * xt.w);
    }
    bvals[0 * NSEG + seg] = ysp;
    bvals[1 * NSEG + seg] = ylp;
    bvals[2 * NSEG + seg] = yst;
    bvals[3 * NSEG + seg] = ylt;
}

// ---------------------------------------------------------------------------
// Phase 2: serial carry propagation across segments within each row.
// 4 scans x 32 rows = 128 independent chains of SPR steps:
//   carry[k] = y ; y = B[k] + A * y
// Loads are independent of the FMA chain, so they pipeline.
// ---------------------------------------------------------------------------
__global__ void __launch_bounds__(128)
ldr_phase2_carry(const float* __restrict__ bvals,
                 float* __restrict__ carries,   // [4][NSEG]
                 float As, float Al)
{
    const int tid  = threadIdx.x;          // 0..127
    const int scan = tid >> 5;             // 0..3
    const int row  = tid & 31;             // 0..31
    const float A  = (scan & 1) ? Al : As; // odd scans are "long"
    const float* __restrict__ bv = bvals   + (size_t)scan * NSEG + (size_t)row * SPR;
    float*       __restrict__ cr = carries + (size_t)scan * NSEG + (size_t)row * SPR;
    float y = 0.f;
    for (int k = 0; k < SPR; ++k) {
        cr[k] = y;
        y = fmaf(A, y, bv[k]);
    }
}

// ---------------------------------------------------------------------------
// Phase 3: re-scan each segment with its exact carry as initial state and
// emit log-domain differences:  d[i] = log(sp/st),  e[i] = log(lp/lt).
// One fast log per pair instead of two (log(a)-log(b) = log(a/b), a,b > 0).
// ---------------------------------------------------------------------------
__global__ void __launch_bounds__(256)
ldr_phase3_rescan(const float* __restrict__ pred,
                  const float* __restrict__ targ,
                  const float* __restrict__ carries,
                  float* __restrict__ d_arr,
                  float* __restrict__ e_arr,
                  float cs, float cms, float cl, float cml)
{
    const int seg = blockIdx.x * blockDim.x + threadIdx.x;
    const int row = seg / SPR;
    const int s   = seg % SPR;
    const size_t base = (size_t)row * T_LEN + (size_t)s * SEG;

    const float4* __restrict__ p4 = reinterpret_cast<const float4*>(pred + base);
    const float4* __restrict__ t4 = reinterpret_cast<const float4*>(targ + base);
    float4* __restrict__ d4 = reinterpret_cast<float4*>(d_arr + base);
    float4* __restrict__ e4 = reinterpret_cast<float4*>(e_arr + base);

    float ysp = carries[0 * NSEG + seg];
    float ylp = carries[1 * NSEG + seg];
    float yst = carries[2 * NSEG + seg];
    float ylt = carries[3 * NSEG + seg];

#pragma unroll 2
    for (int i = 0; i < SEG / 4; ++i) {
        float4 xp = p4[i];
        float4 xt = t4[i];
        float4 dd, ee;
        ysp = fmaf(cms, ysp, cs * xp.x); yst = fmaf(cms, yst, cs * xt.x);
        ylp = fmaf(cml, ylp, cl * xp.x); ylt = fmaf(cml, ylt, cl * xt.x);
        dd.x = __logf(ysp / yst); ee.x = __logf(ylp / ylt);
        ysp = fmaf(cms, ysp, cs * xp.y); yst = fmaf(cms, yst, cs * xt.y);
        ylp = fmaf(cml, ylp, cl * xp.y); ylt = fmaf(cml, ylt, cl * xt.y);
        dd.y = __logf(ysp / yst); ee.y = __logf(ylp / ylt);
        ysp = fmaf(cms, ysp, cs * xp.z); yst = fmaf(cms, yst, cs * xt.z);
        ylp = fmaf(cml, ylp, cl * xp.z); ylt = fmaf(cml, ylt, cl * xt.z);
        dd.z = __logf(ysp / yst); ee.z = __logf(ylp / ylt);
        ysp = fmaf(cms, ysp, cs * xp.w); yst = fmaf(cms, yst, cs * xt.w);
        ylp = fmaf(cml, ylp, cl * xp.w); ylt = fmaf(cml, ylt, cl * xt.w);
        dd.w = __logf(ysp / yst); ee.w = __logf(ylp / ylt);
        d4[i] = dd;
        e4[i] = ee;
    }
}

// ---------------------------------------------------------------------------
// Phase 4: reduction of mean|d[i] - e[(i+SHIFT) mod N]|.
// Sized tail-free: 1024 blocks x 256 threads x 32 iters == 2^23 == N_TOT,
// so EXEC stays all-ones through the WMMA (ISA requirement).
// Wave reduction uses V_WMMA_F32_16X16X4_F32 with a ones B-matrix:
//   A[m][0]=p(lane m), A[m][2]=p(lane 16+m) -> D[m][*] = p[m]+p[m+16];
// summing a lane's 8 D-VGPRs gives the half-wave row sums, and one
// shfl_xor(16) completes the full-wave sum (f32 end to end, no precision loss).
// ---------------------------------------------------------------------------
__global__ void __launch_bounds__(256)
ldr_phase4_reduce(const float* __restrict__ d_arr,
                  const float* __restrict__ e_arr,
                  float* __restrict__ partials)   // [1024]
{
    const int gid = blockIdx.x * 256 + threadIdx.x;   // 0..262143
    float p = 0.f;
#pragma unroll 4
    for (int k = 0; k < 32; ++k) {
        int i = gid + k * 262144;
        int j = i + SHIFT;
        if (j >= N_TOT) j -= N_TOT;
        p += fabsf(d_arr[i] - e_arr[j]);
    }

    // Wave-level sum on the matrix unit (f32 WMMA).
    v2f a;  a[0] = p;   a[1] = 0.f;
    v2f b;  b[0] = 1.f; b[1] = 1.f;
    v8f c = {};
    v8f d = __builtin_amdgcn_wmma_f32_16x16x4_f32(
        /*neg_a=*/false, a, /*neg_b=*/false, b,
        /*c_mod=*/(short)0, c, /*reuse_a=*/false, /*reuse_b=*/false);
    float s = d[0] + d[1] + d[2] + d[3] + d[4] + d[5] + d[6] + d[7];
    s += __shfl_xor(s, 16, 32);        // combine the two lane halves

    __shared__ float lds[8];
    const int lane = threadIdx.x & 31;
    const int wave = threadIdx.x >> 5;
    if (lane == 0) lds[wave] = s;
    __syncthreads();
    if (threadIdx.x == 0) {
        float bs = 0.f;
        for (int w = 0; w < 8; ++w) bs += lds[w];   // fixed order: deterministic
        partials[blockIdx.x] = bs;
    }
}

// ---------------------------------------------------------------------------
// Phase 5: deterministic fixed-order final sum of 1024 block partials.
// ---------------------------------------------------------------------------
__global__ void __launch_bounds__(256)
ldr_phase5_final(const float* __restrict__ partials, float* __restrict__ out)
{
    __shared__ float lds[256];
    const int t = threadIdx.x;
    float s = 0.f;
    for (int k = 0; k < 4; ++k) s += partials[t * 4 + k];
    lds[t] = s;
    __syncthreads();
    for (int off = 128; off > 0; off >>= 1) {
        if (t < off) lds[t] += lds[t + off];
        __syncthreads();
    }
    if (t == 0) out[0] = lds[0] * (1.0f / (float)N_TOT);   // N = 2^23, exact
}

// ---------------------------------------------------------------------------
// Host launcher (graph-capture safe: only kernel launches on `stream`).
// ---------------------------------------------------------------------------
extern "C" void kernel_launch(void* const* d_in, const int* in_sizes, int n_in,
                              void* d_out, int out_size, void* d_ws, size_t ws_size,
                              hipStream_t stream)
{
    const float* pred = (const float*)d_in[0];
    const float* targ = (const float*)d_in[1];
    float* out = (float*)d_out;

    float* ws     = (float*)d_ws;
    float* d_arr  = ws;                                   // N_TOT floats
    float* e_arr  = ws + (size_t)N_TOT;                   // N_TOT floats
    float* bvals  = ws + 2 * (size_t)N_TOT;               // 4*NSEG floats
    float* carr   = bvals + 4 * (size_t)NSEG;             // 4*NSEG floats
    float* parts  = carr  + 4 * (size_t)NSEG;             // 1024 floats
    (void)in_sizes; (void)n_in; (void)out_size; (void)ws_size;

    // torchcomp-style coefficients, matching np.float32(1 - exp(-2200/(ms*sr)))
    const double cs_d = 1.0 - exp(-2200.0 / (50.0   * 44100.0));
    const double cl_d = 1.0 - exp(-2200.0 / (3000.0 * 44100.0));
    const float cs  = (float)cs_d;
    const float cl  = (float)cl_d;
    const float cms = 1.0f - cs;                 // matches reference's a = 1-coef
    const float cml = 1.0f - cl;
    const float As  = (float)pow((double)cms, (double)SEG);  // segment decay
    const float Al  = (float)pow((double)cml, (double)SEG);

    ldr_phase1_segscan<<<NSEG / 256, 256, 0, stream>>>(pred, targ, bvals, cs, cms, cl, cml);
    ldr_phase2_carry  <<<1, 128, 0, stream>>>(bvals, carr, As, Al);
    ldr_phase3_rescan <<<NSEG / 256, 256, 0, stream>>>(pred, targ, carr, d_arr, e_arr,
                                                       cs, cms, cl, cml);
    ldr_phase4_reduce <<<1024, 256, 0, stream>>>(d_arr, e_arr, parts);
    ldr_phase5_final  <<<1, 256, 0, stream>>>(parts, out);
}